// HybridQGNN_32152125178145
// MI455X (gfx1250) — compile-verified
//
#include <hip/hip_runtime.h>
#include <cstdint>

typedef float v2f __attribute__((ext_vector_type(2)));
typedef float v8f __attribute__((ext_vector_type(8)));

// ---------------------------------------------------------------- zero scratch
__global__ void zero_f32(float* __restrict__ p, long long n) {
    long long i = (long long)blockIdx.x * blockDim.x + threadIdx.x;
    long long stride = (long long)gridDim.x * blockDim.x;
    for (; i < n; i += stride) p[i] = 0.0f;
}

// ------------------------------------------------- edge gather + scatter-add
// one thread per (edge, channel); 64 channels per edge -> 2 waves/edge, fully
// coalesced reads of feat[src], f32 atomics into agg[dst] (L2-resident).
__global__ void __launch_bounds__(256)
scatter_accum(const float* __restrict__ feat,
              const long long* __restrict__ src,
              const long long* __restrict__ dst,
              float* __restrict__ agg,
              float* __restrict__ cnt,
              long long n_edges) {
    long long t = (long long)blockIdx.x * blockDim.x + threadIdx.x;
    long long total = n_edges << 6;
    if (t >= total) return;
    long long e = t >> 6;
    int c = (int)(t & 63);
    int s = (int)src[e];
    int d = (int)dst[e];
    atomicAdd(&agg[(long long)d * 64 + c], feat[(long long)s * 64 + c]);
    if (c == 0) atomicAdd(&cnt[d], 1.0f);
}

// ---------------------------------------------- fused SAGE linear (WMMA f32)
// out[m,:] = relu( (agg[m,:]/max(cnt,1)) @ Wl^T + bl + x[m,:] @ Wr^T )
// One wave handles a 16-node tile, looping over 4 column tiles of the 64-wide
// output. A-fragment (16x4 f32): lanes 0-15 hold K=k,k+1; lanes 16-31 K=k+2,k+3
// for row M = lane&15.  B-fragment (4x16): same K split, N = lane&15.
__global__ void __launch_bounds__(256)
sage_linear_relu(const float* __restrict__ agg, const float* __restrict__ cnt,
                 const float* __restrict__ x,
                 const float* __restrict__ Wl, const float* __restrict__ bl,
                 const float* __restrict__ Wr,
                 float* __restrict__ out, int n_nodes) {
    int wave = threadIdx.x >> 5;
    int lane = threadIdx.x & 31;
    int tile = blockIdx.x * 8 + wave;
    int node_base = tile * 16;
    if (node_base >= n_nodes) return;   // wave-uniform: EXEC stays all-1s

    int mrow  = lane & 15;
    int row   = node_base + mrow;
    int rowc  = (row < n_nodes) ? row : (n_nodes - 1);
    int khalf = (lane < 16) ? 0 : 2;

    float c0   = cnt[rowc];
    float invc = 1.0f / (c0 > 1.0f ? c0 : 1.0f);
    const float* arow = agg + (long long)rowc * 64;
    const float* xrow = x   + (long long)rowc * 64;

    for (int nt = 0; nt < 4; ++nt) {
        int col = nt * 16 + mrow;       // output column for B/C fragments
        v8f acc = {0.f, 0.f, 0.f, 0.f, 0.f, 0.f, 0.f, 0.f};

        #pragma unroll
        for (int k = 0; k < 64; k += 4) {       // mean-neighbors @ Wl^T
            int kk = k + khalf;
            v2f a, b;
            a.x = arow[kk]     * invc;
            a.y = arow[kk + 1] * invc;
            b.x = Wl[col * 64 + kk];
            b.y = Wl[col * 64 + kk + 1];
            acc = __builtin_amdgcn_wmma_f32_16x16x4_f32(
                      false, a, false, b, (short)0, acc, false, false);
        }
        #pragma unroll
        for (int k = 0; k < 64; k += 4) {       // root features @ Wr^T
            int kk = k + khalf;
            v2f a, b;
            a.x = xrow[kk];
            a.y = xrow[kk + 1];
            b.x = Wr[col * 64 + kk];
            b.y = Wr[col * 64 + kk + 1];
            acc = __builtin_amdgcn_wmma_f32_16x16x4_f32(
                      false, a, false, b, (short)0, acc, false, false);
        }

        float bias = bl[col];
        #pragma unroll
        for (int j = 0; j < 8; ++j) {
            // C/D layout: VGPR j -> M=j (lanes 0-15) / M=j+8 (lanes 16-31)
            int m = node_base + ((lane < 16) ? j : j + 8);
            if (m < n_nodes) {
                float v = acc[j] + bias;
                out[(long long)m * 64 + col] = (v > 0.0f) ? v : 0.0f;
            }
        }
    }
}

// ------------------------------------------- compressor + QNN surrogate head
// out[n] = sigmoid( Wcls * prod_j cos( h[n,:]@Wc[j] + bc[j] + qw[j] ) + bcls )
__global__ void __launch_bounds__(128)
qnn_head(const float* __restrict__ h, const float* __restrict__ Wc,
         const float* __restrict__ bc, const float* __restrict__ qw,
         const float* __restrict__ Wcls, const float* __restrict__ bcls,
         float* __restrict__ out, int n_nodes) {
    __shared__ float lds[128 * 65];     // pitch 65 to avoid bank conflicts
    int base = blockIdx.x * 128;
    for (int i = threadIdx.x; i < 128 * 64; i += 128) {
        int r = i >> 6, cc = i & 63;
        int node = base + r;
        lds[r * 65 + cc] = (node < n_nodes) ? h[(long long)node * 64 + cc] : 0.0f;
    }
    __syncthreads();

    int node = base + threadIdx.x;
    if (node >= n_nodes) return;
    const float* rowp = &lds[threadIdx.x * 65];

    float q = 1.0f;
    #pragma unroll
    for (int j = 0; j < 8; ++j) {
        float a = bc[j];
        #pragma unroll
        for (int k = 0; k < 64; ++k) a += rowp[k] * Wc[j * 64 + k];
        q *= __cosf(a + qw[j]);
    }
    float z = q * Wcls[0] + bcls[0];
    out[node] = 1.0f / (1.0f + __expf(-z));
}

// -------------------------------------------------------------------- driver
extern "C" void kernel_launch(void* const* d_in, const int* in_sizes, int n_in,
                              void* d_out, int out_size, void* d_ws, size_t ws_size,
                              hipStream_t stream) {
    const float*      x    = (const float*)d_in[0];
    const long long*  ei   = (const long long*)d_in[1];  // int64 per reference
    const float*      Wl1  = (const float*)d_in[2];
    const float*      bl1  = (const float*)d_in[3];
    const float*      Wr1  = (const float*)d_in[4];
    const float*      Wl2  = (const float*)d_in[5];
    const float*      bl2  = (const float*)d_in[6];
    const float*      Wr2  = (const float*)d_in[7];
    const float*      Wc   = (const float*)d_in[8];
    const float*      bc   = (const float*)d_in[9];
    const float*      qw   = (const float*)d_in[10];
    const float*      Wcls = (const float*)d_in[11];
    const float*      bcls = (const float*)d_in[12];
    float*            out  = (float*)d_out;

    long long N = in_sizes[0] / 64;
    long long E = in_sizes[1] / 2;
    const long long* src = ei;
    const long long* dst = ei + E;

    float* agg = (float*)d_ws;          // N*64
    float* cnt = agg + N * 64;          // N
    float* h1  = cnt + N;               // N*64
    float* h2  = h1 + N * 64;           // N*64

    long long zn = N * 64 + N;          // agg + cnt are contiguous
    int zgrid = (int)((zn + 255) / 256);
    long long sthreads = E * 64;
    int sgrid = (int)((sthreads + 255) / 256);
    int ltiles = (int)((N + 15) / 16);
    int lgrid  = (ltiles + 7) / 8;
    int hgrid  = (int)((N + 127) / 128);

    // ---- layer 1
    zero_f32<<<zgrid, 256, 0, stream>>>(agg, zn);
    scatter_accum<<<sgrid, 256, 0, stream>>>(x, src, dst, agg, cnt, E);
    sage_linear_relu<<<lgrid, 256, 0, stream>>>(agg, cnt, x, Wl1, bl1, Wr1, h1, (int)N);
    // ---- layer 2
    zero_f32<<<zgrid, 256, 0, stream>>>(agg, zn);
    scatter_accum<<<sgrid, 256, 0, stream>>>(h1, src, dst, agg, cnt, E);
    sage_linear_relu<<<lgrid, 256, 0, stream>>>(agg, cnt, h1, Wl2, bl2, Wr2, h2, (int)N);
    // ---- compressor + quantum surrogate + classifier
    qnn_head<<<hgrid, 128, 0, stream>>>(h2, Wc, bc, qw, Wcls, bcls, out, (int)N);
}